// Multi_ATTN_9199819948231
// MI455X (gfx1250) — compile-verified
//
#include <hip/hip_runtime.h>
#include <math.h>

typedef __attribute__((ext_vector_type(2))) float v2f;
typedef __attribute__((ext_vector_type(8))) float v8f;

#define Bsz 8
#define LEN 512
#define DIM 512
#define NH  8
#define HD  64
#define NBLK 18
#define PER_BH (LEN * LEN)          // 262144
#define PER_T  (Bsz * LEN * DIM)    // 2097152 floats per activation tensor
#define FEATW  3072                 // 6 * DIM pooled feature columns

#define WMMA_F32(av, bv, cv) \
    __builtin_amdgcn_wmma_f32_16x16x4_f32(false, (av), false, (bv), (short)0, (cv), false, false)

// ---------------------------------------------------------------------------
// Generic strided, 2-level-batched GEMM on V_WMMA_F32_16X16X4_F32.
// One wave computes a 32(M) x 64(N) strip of C: 8 accumulators, 2 A fragments
// reused across 4 N tiles, 4 B fragments reused across 2 M tiles
// (12 dword loads per 8 WMMAs ~ 10.7 FLOP/B of fragment traffic).
// K-loop is pure pointer bumps; the K%4 tail (only K=205) is one guarded step
// so the hot loop has no exec-mask divergence (WMMA needs EXEC all-ones).
//   A(m,k)=A[m*sAm+k*sAk], B(k,n)=B[k*sBk+n*sBn], C(m,n)=C[m*sCm+n]
//   batch z: offset = (z/nInner)*bat?1 + (z%nInner)*bat?2
//   C = alpha*A@B + beta*C   (beta read from cPtr[cIdx] if cPtr != null)
//   optional epilogue: C -= 1e8*(1 - mask[(z/maskDiv)*N + n])
// VGPR layouts per ISA 7.12.2:
//   A 16x4 : lane L -> M=L&15, K = 2*(L>>4)+v ;  B 4x16 : lane L -> N=L&15
//   C 16x16: vgpr r, lane L -> M = r + 8*(L>>4), N = L&15
// ---------------------------------------------------------------------------
__global__ void wmma_gemm_f32(
    const float* __restrict__ A, const float* __restrict__ B,
    float* __restrict__ C,
    int M, int N, int K,
    int sAm, int sAk, int sBk, int sBn, int sCm,
    long long batA1, long long batA2,
    long long batB1, long long batB2,
    long long batC1, long long batC2, int nInner,
    float alpha, float betaImm, const float* __restrict__ cPtr, int cIdx,
    const float* __restrict__ maskPtr, int maskDiv)
{
    const int lane = threadIdx.x;        // 0..31
    const int half = lane >> 4;          // 0/1
    const int l16  = lane & 15;
    const int z    = blockIdx.z;
    const int zo   = z / nInner;
    const int zi   = z - zo * nInner;

    const float* Ab = A + (long long)zo * batA1 + (long long)zi * batA2;
    const float* Bb = B + (long long)zo * batB1 + (long long)zi * batB2;
    float*       Cb = C + (long long)zo * batC1 + (long long)zi * batC2;

    const int tM = blockIdx.y << 5;      // 32-tall M strip
    const int tN = blockIdx.x << 6;      // 64-wide N strip

    const float beta = cPtr ? cPtr[cIdx] : betaImm;

    // per-lane streaming pointers (K advances by pointer bump only)
    const float* ap0 = Ab + (long long)(tM + l16) * sAm + (long long)(2 * half) * sAk;
    const float* ap1 = ap0 + 16LL * sAm;
    const float* bp  = Bb + (long long)(tN + l16) * sBn + (long long)(2 * half) * sBk;
    const long long bT = 16LL * sBn;     // next 16-wide N tile
    const long long aK = sAk;
    const long long bK = sBk;

    v8f acc[2][4];
#pragma unroll
    for (int mi = 0; mi < 2; ++mi)
#pragma unroll
        for (int t = 0; t < 4; ++t) { v8f zf = {}; acc[mi][t] = zf; }

    const int K4 = K & ~3;
#pragma unroll 2
    for (int k = 0; k < K4; k += 4) {
        v2f a0, a1, bf[4];
        a0.x = ap0[0];  a0.y = ap0[aK];
        a1.x = ap1[0];  a1.y = ap1[aK];
#pragma unroll
        for (int t = 0; t < 4; ++t) {
            bf[t].x = bp[t * bT];
            bf[t].y = bp[t * bT + bK];
        }
#pragma unroll
        for (int t = 0; t < 4; ++t) {
            acc[0][t] = WMMA_F32(a0, bf[t], acc[0][t]);
            acc[1][t] = WMMA_F32(a1, bf[t], acc[1][t]);
        }
        ap0 += 4LL * sAk;
        ap1 += 4LL * sAk;
        bp  += 4LL * sBk;
    }
    if (K4 < K) {                         // K remainder 1..3 (only K=205 path)
        const int ka = K4 + 2 * half;
        const bool e0 = (ka < K), e1 = (ka + 1 < K);
        v2f a0, a1, bf[4];
        a0.x = e0 ? ap0[0]  : 0.0f;  a0.y = e1 ? ap0[aK] : 0.0f;
        a1.x = e0 ? ap1[0]  : 0.0f;  a1.y = e1 ? ap1[aK] : 0.0f;
#pragma unroll
        for (int t = 0; t < 4; ++t) {
            bf[t].x = e0 ? bp[t * bT]      : 0.0f;
            bf[t].y = e1 ? bp[t * bT + bK] : 0.0f;
        }
#pragma unroll
        for (int t = 0; t < 4; ++t) {
            acc[0][t] = WMMA_F32(a0, bf[t], acc[0][t]);
            acc[1][t] = WMMA_F32(a1, bf[t], acc[1][t]);
        }
    }

#pragma unroll
    for (int mi = 0; mi < 2; ++mi) {
#pragma unroll
        for (int t = 0; t < 4; ++t) {
            const int n = tN + 16 * t + l16;
#pragma unroll
            for (int r = 0; r < 8; ++r) {
                const int m = tM + 16 * mi + r + 8 * half;
                float val = alpha * acc[mi][t][r];
                float* cp = Cb + (long long)m * sCm + n;
                if (beta != 0.0f) val += beta * (*cp);
                if (maskPtr)      val -= 1e8f * (1.0f - maskPtr[(long long)(z / maskDiv) * N + n]);
                *cp = val;
            }
        }
    }
}

// ---------------------------------------------------------------------------
// Row-wise LayerNorm in place.
// ---------------------------------------------------------------------------
__global__ void layernorm_rows(float* __restrict__ X,
                               const float* __restrict__ g,
                               const float* __restrict__ b, int width)
{
    __shared__ float red[256];
    float* row = X + (long long)blockIdx.x * width;
    const int t = threadIdx.x;

    float s = 0.0f;
    for (int d = t; d < width; d += 256) s += row[d];
    red[t] = s; __syncthreads();
    for (int o = 128; o > 0; o >>= 1) { if (t < o) red[t] += red[t + o]; __syncthreads(); }
    const float mu = red[0] / (float)width;
    __syncthreads();

    float v = 0.0f;
    for (int d = t; d < width; d += 256) { float dv = row[d] - mu; v += dv * dv; }
    red[t] = v; __syncthreads();
    for (int o = 128; o > 0; o >>= 1) { if (t < o) red[t] += red[t + o]; __syncthreads(); }
    const float inv = rsqrtf(red[0] / (float)width + 1e-5f);

    for (int d = t; d < width; d += 256)
        row[d] = (row[d] - mu) * inv * g[d] + b[d];
}

// ---------------------------------------------------------------------------
// Row-wise softmax: A = softmax(S).
// ---------------------------------------------------------------------------
__global__ void softmax_rows(const float* __restrict__ S,
                             float* __restrict__ A, int width)
{
    __shared__ float red[256];
    const float* row = S + (long long)blockIdx.x * width;
    float* arow      = A + (long long)blockIdx.x * width;
    const int t = threadIdx.x;

    float m = -INFINITY;
    for (int d = t; d < width; d += 256) m = fmaxf(m, row[d]);
    red[t] = m; __syncthreads();
    for (int o = 128; o > 0; o >>= 1) { if (t < o) red[t] = fmaxf(red[t], red[t + o]); __syncthreads(); }
    m = red[0]; __syncthreads();

    float s = 0.0f;
    for (int d = t; d < width; d += 256) { float e = expf(row[d] - m); arow[d] = e; s += e; }
    red[t] = s; __syncthreads();
    for (int o = 128; o > 0; o >>= 1) { if (t < o) red[t] += red[t + o]; __syncthreads(); }
    const float inv = 1.0f / red[0];

    for (int d = t; d < width; d += 256) arow[d] *= inv;
}

// ---------------------------------------------------------------------------
// Column sum/max pooling accumulation. grid=(B, 2), block=256.
// ---------------------------------------------------------------------------
__global__ void pool_accum(const float* __restrict__ Y,
                           float* __restrict__ fsum, float* __restrict__ fmax,
                           int colOff)
{
    const int b = blockIdx.x;
    const int d = blockIdx.y * 256 + threadIdx.x;     // 0..511
    const float* p = Y + (long long)b * (LEN * DIM) + d;
    float s = 0.0f, m = -INFINITY;
    for (int r = 0; r < LEN; ++r) {
        float v = p[(long long)r * DIM];
        s += v; m = fmaxf(m, v);
    }
    const int idx = b * FEATW + colOff + d;
    fsum[idx] += s;
    fmax[idx] = fmaxf(fmax[idx], m);
}

__global__ void init_feats(float* fsum, float* fmax, int n)
{
    int i = blockIdx.x * 256 + threadIdx.x;
    if (i < n) { fsum[i] = 0.0f; fmax[i] = -INFINITY; }
}

// ---------------------------------------------------------------------------
// out[b,c] = [mean | max] feats (6144) @ clf_w (6144 x 9). grid=(9, B).
// ---------------------------------------------------------------------------
__global__ void classifier(const float* __restrict__ fsum,
                           const float* __restrict__ fmax,
                           const float* __restrict__ W,
                           float* __restrict__ out)
{
    __shared__ float red[256];
    const int cc = blockIdx.x;
    const int b  = blockIdx.y;
    const int t  = threadIdx.x;
    float acc = 0.0f;
    for (int f = t; f < 2 * FEATW; f += 256) {
        float feat = (f < FEATW) ? fsum[b * FEATW + f] * (1.0f / 1536.0f)
                                 : fmax[b * FEATW + (f - FEATW)];
        acc += feat * W[f * 9 + cc];
    }
    red[t] = acc; __syncthreads();
    for (int o = 128; o > 0; o >>= 1) { if (t < o) red[t] += red[t + o]; __syncthreads(); }
    if (t == 0) out[b * 9 + cc] = red[0];
}

// ---------------------------------------------------------------------------
// Host side
// ---------------------------------------------------------------------------
static void gemm(hipStream_t st, const float* A, const float* B, float* C,
                 int M, int N, int K,
                 int sAm, int sAk, int sBk, int sBn, int sCm,
                 long long bA1, long long bA2, long long bB1, long long bB2,
                 long long bC1, long long bC2, int nInner, int nBatch,
                 float alpha, float betaImm, const float* cPtr, int cIdx,
                 const float* maskPtr, int maskDiv)
{
    dim3 grid(N / 64, M / 32, nBatch);
    wmma_gemm_f32<<<grid, dim3(32), 0, st>>>(
        A, B, C, M, N, K, sAm, sAk, sBk, sBn, sCm,
        bA1, bA2, bB1, bB2, bC1, bC2, nInner,
        alpha, betaImm, cPtr, cIdx, maskPtr, maskDiv);
}

extern "C" void kernel_launch(void* const* d_in, const int* in_sizes, int n_in,
                              void* d_out, int out_size, void* d_ws, size_t ws_size,
                              hipStream_t stream)
{
    const float* l      = (const float*)d_in[0];
    const float* v      = (const float*)d_in[1];
    const float* a      = (const float*)d_in[2];
    const float* l_mask = (const float*)d_in[3];
    const float* v_mask = (const float*)d_in[4];
    const float* a_mask = (const float*)d_in[5];
    const float* wl     = (const float*)d_in[6];
    const float* wv     = (const float*)d_in[7];
    const float* wa     = (const float*)d_in[8];
    const float* n1_g   = (const float*)d_in[9];
    const float* n1_b   = (const float*)d_in[10];
    const float* proj_w = (const float*)d_in[11];
    const float* minus_w= (const float*)d_in[12];
    const float* n2_g   = (const float*)d_in[13];
    const float* n2_b   = (const float*)d_in[14];
    const float* cvec   = (const float*)d_in[15];
    const float* clf_w  = (const float*)d_in[16];
    float* out = (float*)d_out;

    float* w   = (float*)d_ws;
    float* lu  = w;
    float* vu  = w + (long long)1 * PER_T;
    float* au  = w + (long long)2 * PER_T;
    float* qA  = w + (long long)3 * PER_T;
    float* qB  = w + (long long)4 * PER_T;
    float* ctx = w + (long long)5 * PER_T;
    float* xb  = w + (long long)6 * PER_T;
    float* sc  = w + (long long)7 * PER_T;    // scores: 64 * 512*512 = 8*PER_T
    float* att = w + (long long)15 * PER_T;   // 8*PER_T
    float* fsum = w + (long long)23 * PER_T;
    float* fmax = fsum + Bsz * FEATW;

    init_feats<<<(Bsz * FEATW + 255) / 256, 256, 0, stream>>>(fsum, fmax, Bsz * FEATW);

    // ---- Unify_Dimension: U = LN(X @ W) ----
    const int MR = Bsz * LEN;  // 4096 rows
    gemm(stream, l, wl, lu, MR, DIM, 768, 768, 1, DIM, 1, DIM,
         0,0,0,0,0,0, 1, 1, 1.0f, 0.0f, nullptr, 0, nullptr, 1);
    layernorm_rows<<<MR, 256, 0, stream>>>(lu, n1_g, n1_b, DIM);
    gemm(stream, v, wv, vu, MR, DIM, 640, 640, 1, DIM, 1, DIM,
         0,0,0,0,0,0, 1, 1, 1.0f, 0.0f, nullptr, 0, nullptr, 1);
    layernorm_rows<<<MR, 256, 0, stream>>>(vu, n1_g, n1_b, DIM);
    gemm(stream, a, wa, au, MR, DIM, 205, 205, 1, DIM, 1, DIM,
         0,0,0,0,0,0, 1, 1, 1.0f, 0.0f, nullptr, 0, nullptr, 1);
    layernorm_rows<<<MR, 256, 0, stream>>>(au, n1_g, n1_b, DIM);

    const float* U[3]  = { lu, vu, au };
    const float* MK[3] = { l_mask, v_mask, a_mask };
    const int qIdx[9] = {0,0,0, 1,1,1, 2,2,2};
    const int kIdx[9] = {0,1,2, 1,0,2, 2,0,1};
    const float inv_sqrt = 0.125f;   // 1/sqrt(64)

    const long long PS = PER_BH;     // 262144

    for (int s = 0; s < 9; ++s) {
        const float* q_in = U[qIdx[s]];
        const float* kv   = U[kIdx[s]];
        const float* mk   = MK[kIdx[s]];
        for (int j = 0; j < 2; ++j) {
            const int i = s * 2 + j;

            // scores = inv_sqrt * Q K^T + c[i]*scores_prev - 1e8*(1-mask)
            gemm(stream, q_in, kv, sc, LEN, LEN, HD,
                 DIM, 1,            // A: q rows, d within head
                 1, DIM,            // B(d,k) = kv[k*DIM + d]
                 LEN,
                 PS, HD,            // A batch: per-b, per-h
                 PS, HD,            // B batch
                 8 * PS, PS,        // C batch: [b][h] 512x512 blocks
                 NH, Bsz * NH,
                 inv_sqrt,
                 0.0f, (j == 0) ? nullptr : cvec, i,
                 mk, NH);

            softmax_rows<<<Bsz * NH * LEN, 256, 0, stream>>>(sc, att, LEN);

            // ctx = att @ K   (per batch-head, N = 64)
            gemm(stream, att, kv, ctx, LEN, HD, LEN,
                 LEN, 1,
                 DIM, 1,            // B(k,d) = kv[k*DIM + d]
                 DIM,
                 8 * PS, PS,
                 PS, HD,
                 PS, HD,
                 NH, Bsz * NH,
                 1.0f, 0.0f, nullptr, 0, nullptr, 1);

            // x = ctx @ proj_w[i]
            gemm(stream, ctx, proj_w + (long long)i * DIM * DIM, xb,
                 MR, DIM, DIM, DIM, 1, DIM, 1, DIM,
                 0,0,0,0,0,0, 1, 1, 1.0f, 0.0f, nullptr, 0, nullptr, 1);

            // y = q_in @ Wm[:512] + x @ Wm[512:]
            float* qo = (j == 0) ? qA : qB;
            const float* Wm = minus_w + (long long)i * 1024 * DIM;
            gemm(stream, q_in, Wm, qo,
                 MR, DIM, DIM, DIM, 1, DIM, 1, DIM,
                 0,0,0,0,0,0, 1, 1, 1.0f, 0.0f, nullptr, 0, nullptr, 1);
            gemm(stream, xb, Wm + (long long)DIM * DIM, qo,
                 MR, DIM, DIM, DIM, 1, DIM, 1, DIM,
                 0,0,0,0,0,0, 1, 1, 1.0f, 1.0f, nullptr, 0, nullptr, 1);

            layernorm_rows<<<MR, 256, 0, stream>>>(qo, n2_g + (long long)i * DIM,
                                                   n2_b + (long long)i * DIM, DIM);

            const int colOff = ((s % 3) * 2 + j) * DIM;
            pool_accum<<<dim3(Bsz, 2), 256, 0, stream>>>(qo, fsum, fmax, colOff);

            q_in = qo;
        }
    }

    classifier<<<dim3(9, Bsz), 256, 0, stream>>>(fsum, fmax, clf_w, out);
}